// HopfBifur_80221399154898
// MI455X (gfx1250) — compile-verified
//
#include <hip/hip_runtime.h>

// Elementwise Dormand-Prince RK5 integration of dz/dt = (a - z^2) z,
// 96 fixed steps over [0, 2*pi], fp32, 4096x4096 independent elements.
//
// Roofline: ~3072 FMA-class ops/element vs 12 B of HBM traffic
// (~500 FLOP/B) -> purely fp32-VALU bound. Peak path on MI455X is
// v_pk_fma_f32 (VOP3P packed fp32 FMA, 2 FMAs/instr) with the h-folded
// Butcher coefficients broadcast from SGPRs. WMMA does not apply (serial
// per-element dependence chains, fp32 precision required).

typedef float v2f __attribute__((ext_vector_type(2)));

// Step size folded into the tableau at compile time (double fold, one f32 round).
#define HC(num, den) \
  ((float)((6.2831853071795864769252867665590 / 96.0) * ((double)(num) / (double)(den))))

__device__ __forceinline__ v2f splat(float c) { v2f r = {c, c}; return r; }

__device__ __forceinline__ v2f rhs(v2f y, v2f a) {
#pragma clang fp contract(fast)
  // (a - y*y) * y  ->  v_pk_fma(-y, y, a) ; v_pk_mul
  return (a - y * y) * y;
}

// One DOPRI5 step with h pre-multiplied into the coefficients (32 packed ops).
__device__ __forceinline__ v2f dopri_step(v2f z, v2f a) {
#pragma clang fp contract(fast)
  const v2f a21 = splat(HC(1, 5));
  const v2f a31 = splat(HC(3, 40)),       a32 = splat(HC(9, 40));
  const v2f a41 = splat(HC(44, 45)),      a42 = splat(HC(-56, 15)),
            a43 = splat(HC(32, 9));
  const v2f a51 = splat(HC(19372, 6561)), a52 = splat(HC(-25360, 2187)),
            a53 = splat(HC(64448, 6561)), a54 = splat(HC(-212, 729));
  const v2f a61 = splat(HC(9017, 3168)),  a62 = splat(HC(-355, 33)),
            a63 = splat(HC(46732, 5247)), a64 = splat(HC(49, 176)),
            a65 = splat(HC(-5103, 18656));
  const v2f b1 = splat(HC(35, 384)),  b3 = splat(HC(500, 1113)),
            b4 = splat(HC(125, 192)), b5 = splat(HC(-2187, 6784)),
            b6 = splat(HC(11, 84));

  v2f k1 = rhs(z, a);
  v2f k2 = rhs(z + a21 * k1, a);
  v2f k3 = rhs(z + a31 * k1 + a32 * k2, a);
  v2f k4 = rhs(z + a41 * k1 + a42 * k2 + a43 * k3, a);
  v2f k5 = rhs(z + a51 * k1 + a52 * k2 + a53 * k3 + a54 * k4, a);
  v2f k6 = rhs(z + a61 * k1 + a62 * k2 + a63 * k3 + a64 * k4 + a65 * k5, a);
  return z + b1 * k1 + b3 * k3 + b4 * k4 + b5 * k5 + b6 * k6;
}

// Hot kernel: exactly 4 elements per thread, no tail logic, branch-free body.
__global__ __launch_bounds__(256) void hopf_dopri96_vec4(
    const float* __restrict__ states, const float* __restrict__ alphas,
    float* __restrict__ out, int n4) {
  int t = blockIdx.x * blockDim.x + threadIdx.x;
  if (t >= n4) return;
  size_t base = (size_t)t * 4;

  const float4 zv = *(const float4*)(states + base);  // global_load_b128
  const float4 av = *(const float4*)(alphas + base);
  v2f z0 = {zv.x, zv.y}, z1 = {zv.z, zv.w};
  v2f a0 = {av.x, av.y}, a1 = {av.z, av.w};
#pragma unroll 1
  for (int s = 0; s < 96; ++s) {
    z0 = dopri_step(z0, a0);  // two packed streams -> 4 independent chains
    z1 = dopri_step(z1, a1);
  }
  float4 r = {z0.x, z0.y, z1.x, z1.y};
  *(float4*)(out + base) = r;  // global_store_b128
}

// Remainder kernel (only launched when n % 4 != 0; never for 4096x4096).
__global__ __launch_bounds__(64) void hopf_dopri96_tail(
    const float* __restrict__ states, const float* __restrict__ alphas,
    float* __restrict__ out, int start, int n) {
  int i = start + blockIdx.x * blockDim.x + threadIdx.x;
  if (i >= n) return;
  v2f z = splat(states[i]);
  v2f a = splat(alphas[i]);
#pragma unroll 1
  for (int s = 0; s < 96; ++s) z = dopri_step(z, a);
  out[i] = z.x;
}

extern "C" void kernel_launch(void* const* d_in, const int* in_sizes, int n_in,
                              void* d_out, int out_size, void* d_ws, size_t ws_size,
                              hipStream_t stream) {
  const float* states = (const float*)d_in[0];
  const float* alphas = (const float*)d_in[1];
  float* out = (float*)d_out;
  int n = in_sizes[0];
  int n4 = n / 4;  // full float4 groups
  if (n4 > 0) {
    const int block = 256;
    int grid = (n4 + block - 1) / block;
    hopf_dopri96_vec4<<<grid, block, 0, stream>>>(states, alphas, out, n4);
  }
  int rem = n - n4 * 4;
  if (rem > 0) {
    hopf_dopri96_tail<<<1, 64, 0, stream>>>(states, alphas, out, n4 * 4, n);
  }
}